// SE2Descriptor_10187662426893
// MI455X (gfx1250) — compile-verified
//
#include <hip/hip_runtime.h>
#include <math.h>

#define NATOMS   60000
#define NENV     600000
#define NEDGE    600000
#define RS_C     3.0f
#define RC_C     4.0f

typedef __attribute__((ext_vector_type(2))) float v2f;
typedef __attribute__((ext_vector_type(8))) float v8f;

// wave-private LDS staging: hardware keeps DS ops in order per wave; this
// waits for outstanding DS and stops the compiler reordering across it.
#define WAVE_SYNC() asm volatile("s_wait_dscnt 0" ::: "memory")

__global__ void zero_kernel(float* __restrict__ p, long n) {
    long i = (long)blockIdx.x * blockDim.x + threadIdx.x;
    if (i < n) p[i] = 0.0f;
}

// ---------------------------------------------------------------------------
// Kernel 1: per-env MLP (WMMA f32) + outer product + atomic scatter.
// 256 threads = 8 waves; each wave owns 16 envs (M=16 tile).
// ---------------------------------------------------------------------------
__launch_bounds__(256)
__global__ void env_kernel(const float* __restrict__ env_vectors,
                           const float* __restrict__ atom_attr,
                           const int*   __restrict__ env_index,
                           const float* __restrict__ W1, const float* __restrict__ b1,
                           const float* __restrict__ W2, const float* __restrict__ b2,
                           const float* __restrict__ W3, const float* __restrict__ b3,
                           float* __restrict__ summed, float* __restrict__ counts) {
    // weights pair-packed: element (k,n) at float index (k/2)*128 + n*2 + (k&1)
    __shared__ float lds_w1[6 * 128];          // K padded 9 -> 12
    __shared__ float lds_w2[32 * 128];
    __shared__ float lds_w3[32 * 128];
    __shared__ float lds_b1[64], lds_b2[64], lds_b3[64];
    __shared__ float lds_h[8][16 * 64];        // per-wave activation stage [16 rows][64]
    __shared__ float lds_dir[8][16 * 4];       // per-wave direct vectors
    __shared__ int   lds_src[8][16];           // per-wave center indices (-1 = invalid)

    const int tid  = threadIdx.x;
    const int wid  = tid >> 5;
    const int lane = tid & 31;

    for (int i = tid; i < 64; i += 256) {
        lds_b1[i] = b1[i]; lds_b2[i] = b2[i]; lds_b3[i] = b3[i];
    }
    for (int i = tid; i < 12 * 64; i += 256) {
        int k = i >> 6, n = i & 63;
        lds_w1[(k >> 1) * 128 + n * 2 + (k & 1)] = (k < 9) ? W1[k * 64 + n] : 0.0f;
    }
    for (int i = tid; i < 64 * 64; i += 256) {
        int k = i >> 6, n = i & 63;
        lds_w2[(k >> 1) * 128 + n * 2 + (k & 1)] = W2[i];
        lds_w3[(k >> 1) * 128 + n * 2 + (k & 1)] = W3[i];
    }
    __syncthreads();

    float* hst = &lds_h[wid][0];
    float* dir = &lds_dir[wid][0];
    const long envBase = (long)blockIdx.x * 128 + (long)wid * 16;

    // -------- stage env scalars + layer-1 input rows (lanes 0..15, 1 env each)
    if (lane < 16) {
        const int  m = lane;
        const long e = envBase + m;
        int   isrc = -1;
        float s = 0.0f, dx = 0.0f, dy = 0.0f, dz = 0.0f;
        float x[12];
        #pragma unroll
        for (int j = 0; j < 12; ++j) x[j] = 0.0f;
        if (e < (long)NENV) {
            float vx = env_vectors[e * 3 + 0];
            float vy = env_vectors[e * 3 + 1];
            float vz = env_vectors[e * 3 + 2];
            float r  = sqrtf(vx * vx + vy * vy + vz * vz);
            float sn;
            if (r < RS_C) {
                sn = 1.0f / r;
            } else if (r < RC_C) {
                float xx = (r - RS_C) / (RC_C - RS_C);
                sn = (1.0f / r) * (xx * xx * xx * (-10.0f + xx * (15.0f - 6.0f * xx)) + 1.0f);
            } else {
                sn = 0.0f;
            }
            s = sn;
            float rinv = 1.0f / r;
            dx = sn * vx * rinv; dy = sn * vy * rinv; dz = sn * vz * rinv;
            isrc = env_index[e];
            int inbr = env_index[NENV + e];
            x[0] = sn;
            #pragma unroll
            for (int j = 0; j < 4; ++j) {
                x[1 + j] = atom_attr[isrc * 4 + j];
                x[5 + j] = atom_attr[inbr * 4 + j];
            }
            atomicAdd(&counts[isrc], 1.0f);
        }
        dir[m * 4 + 0] = s;  dir[m * 4 + 1] = dx;
        dir[m * 4 + 2] = dy; dir[m * 4 + 3] = dz;
        lds_src[wid][m] = isrc;
        #pragma unroll
        for (int j = 0; j < 12; ++j) hst[m * 64 + j] = x[j];
    }
    WAVE_SYNC();

    const int mrow = lane & 15;       // A/B: M or N position within 16
    const int hi   = lane >> 4;       // half-wave selector
    const int koff = hi * 2;          // A K sub-offset for lanes 16..31

    // -------- layer 1: [16x12] @ [12x64], K padded to 12 (3 WMMA steps)
    v8f acc[4] = {};
    #pragma unroll
    for (int k0 = 0; k0 < 12; k0 += 4) {
        v2f a = *(const v2f*)&hst[mrow * 64 + k0 + koff];
        #pragma unroll
        for (int t = 0; t < 4; ++t) {
            v2f b = *(const v2f*)&lds_w1[((k0 >> 1) + hi) * 128 + (t * 16 + mrow) * 2];
            acc[t] = __builtin_amdgcn_wmma_f32_16x16x4_f32(false, a, false, b,
                                                           (short)0, acc[t], false, false);
        }
    }
    WAVE_SYNC();
    #pragma unroll
    for (int t = 0; t < 4; ++t) {
        int n = t * 16 + mrow;
        float bb = lds_b1[n];
        #pragma unroll
        for (int r = 0; r < 8; ++r) {
            int mm = r + hi * 8;
            hst[mm * 64 + n] = tanhf(acc[t][r] + bb);
        }
    }
    WAVE_SYNC();

    // -------- layer 2: [16x64] @ [64x64]
    v8f acc2[4] = {};
    for (int k0 = 0; k0 < 64; k0 += 4) {
        v2f a = *(const v2f*)&hst[mrow * 64 + k0 + koff];
        #pragma unroll
        for (int t = 0; t < 4; ++t) {
            v2f b = *(const v2f*)&lds_w2[((k0 >> 1) + hi) * 128 + (t * 16 + mrow) * 2];
            acc2[t] = __builtin_amdgcn_wmma_f32_16x16x4_f32(false, a, false, b,
                                                            (short)0, acc2[t], false, false);
        }
    }
    WAVE_SYNC();
    #pragma unroll
    for (int t = 0; t < 4; ++t) {
        int n = t * 16 + mrow;
        float bb = lds_b2[n];
        #pragma unroll
        for (int r = 0; r < 8; ++r) {
            int mm = r + hi * 8;
            hst[mm * 64 + n] = tanhf(acc2[t][r] + bb);
        }
    }
    WAVE_SYNC();

    // -------- layer 3 (no tanh): emb stays in C registers
    v8f acc3[4] = {};
    for (int k0 = 0; k0 < 64; k0 += 4) {
        v2f a = *(const v2f*)&hst[mrow * 64 + k0 + koff];
        #pragma unroll
        for (int t = 0; t < 4; ++t) {
            v2f b = *(const v2f*)&lds_w3[((k0 >> 1) + hi) * 128 + (t * 16 + mrow) * 2];
            acc3[t] = __builtin_amdgcn_wmma_f32_16x16x4_f32(false, a, false, b,
                                                            (short)0, acc3[t], false, false);
        }
    }

    // -------- emb (x) direct outer product, scatter-add into summed[src]
    #pragma unroll
    for (int t = 0; t < 4; ++t) {
        int n = t * 16 + mrow;
        float bb = lds_b3[n];
        #pragma unroll
        for (int r = 0; r < 8; ++r) {
            int mm  = r + hi * 8;
            int src = lds_src[wid][mm];
            if (src >= 0) {
                float em = acc3[t][r] + bb;
                float* dst = &summed[((size_t)src * 64 + n) * 4];
                atomicAdd(&dst[0], em * dir[mm * 4 + 0]);
                atomicAdd(&dst[1], em * dir[mm * 4 + 1]);
                atomicAdd(&dst[2], em * dir[mm * 4 + 2]);
                atomicAdd(&dst[3], em * dir[mm * 4 + 3]);
            }
        }
    }
}

// ---------------------------------------------------------------------------
// Kernel 2: per-atom  aggr(64x4) @ aggr[0:8]^T  + mean-sub + L2 normalize.
// One wave per atom; each lane owns 2 rows (d = 2*lane, 2*lane+1).
// ---------------------------------------------------------------------------
__launch_bounds__(256)
__global__ void atom_kernel(const float* __restrict__ summed,
                            const float* __restrict__ counts,
                            float* __restrict__ g) {
    __shared__ float aggrE[8][32];     // per-wave rows 0..7 of aggr
    const int wid  = threadIdx.x >> 5;
    const int lane = threadIdx.x & 31;
    const int i    = blockIdx.x * 8 + wid;
    if (i >= NATOMS) return;

    const float inv = 1.0f / fmaxf(counts[i], 1.0f);
    const float* s = summed + (size_t)i * 256;
    const int d0 = lane * 2, d1 = lane * 2 + 1;
    float a0[4], a1[4];
    #pragma unroll
    for (int a = 0; a < 4; ++a) {
        a0[a] = s[d0 * 4 + a] * inv;
        a1[a] = s[d1 * 4 + a] * inv;
    }
    if (lane < 4) {
        #pragma unroll
        for (int a = 0; a < 4; ++a) {
            aggrE[wid][d0 * 4 + a] = a0[a];
            aggrE[wid][d1 * 4 + a] = a1[a];
        }
    }
    WAVE_SYNC();

    float u[16];
    #pragma unroll
    for (int e = 0; e < 8; ++e) {
        float e0 = aggrE[wid][e * 4 + 0], e1 = aggrE[wid][e * 4 + 1];
        float e2 = aggrE[wid][e * 4 + 2], e3 = aggrE[wid][e * 4 + 3];
        u[e]     = a0[0] * e0 + a0[1] * e1 + a0[2] * e2 + a0[3] * e3;
        u[8 + e] = a1[0] * e0 + a1[1] * e1 + a1[2] * e2 + a1[3] * e3;
    }

    float part = 0.0f;
    #pragma unroll
    for (int j = 0; j < 16; ++j) part += u[j];
    #pragma unroll
    for (int m = 16; m >= 1; m >>= 1) part += __shfl_xor(part, m, 32);
    const float mean = part * (1.0f / 512.0f);

    float ss = 0.0f;
    #pragma unroll
    for (int j = 0; j < 16; ++j) { u[j] -= mean; ss += u[j] * u[j]; }
    #pragma unroll
    for (int m = 16; m >= 1; m >>= 1) ss += __shfl_xor(ss, m, 32);
    const float rn = 1.0f / sqrtf(ss);

    float* go = g + (size_t)i * 512;
    #pragma unroll
    for (int e = 0; e < 8; ++e) {
        go[d0 * 8 + e] = u[e] * rn;
        go[d1 * 8 + e] = u[8 + e] * rn;
    }
}

// ---------------------------------------------------------------------------
// Kernel 3: edge gather  (bandwidth-bound: 1.23 GB streamed out; g is L2-hot)
// ---------------------------------------------------------------------------
__launch_bounds__(256)
__global__ void edge_kernel(const float* __restrict__ g,
                            const int*   __restrict__ edge_index,
                            const float* __restrict__ edge_length,
                            float* __restrict__ out_edge) {
    const int e  = blockIdx.x;
    const int i0 = edge_index[e];
    const int i1 = edge_index[NEDGE + e];
    const float* g0 = g + (size_t)i0 * 512;
    const float* g1 = g + (size_t)i1 * 512;
    float* o = out_edge + (size_t)e * 513;
    const int t = threadIdx.x;
    o[t]       = g0[t]       + g1[t];
    o[t + 256] = g0[t + 256] + g1[t + 256];
    if (t == 0) o[512] = 1.0f / edge_length[e];
}

extern "C" void kernel_launch(void* const* d_in, const int* in_sizes, int n_in,
                              void* d_out, int out_size, void* d_ws, size_t ws_size,
                              hipStream_t stream) {
    const float* env_vectors = (const float*)d_in[0];
    const float* atom_attr   = (const float*)d_in[1];
    const int*   env_index   = (const int*)d_in[2];
    const int*   edge_index  = (const int*)d_in[3];
    const float* edge_length = (const float*)d_in[4];
    const float* W1 = (const float*)d_in[5];
    const float* b1 = (const float*)d_in[6];
    const float* W2 = (const float*)d_in[7];
    const float* b2 = (const float*)d_in[8];
    const float* W3 = (const float*)d_in[9];
    const float* b3 = (const float*)d_in[10];

    float* out      = (float*)d_out;
    float* g        = out;                               // [N, 512]
    float* out_edge = out + (size_t)NATOMS * 512;        // [NEDGE, 513]

    const long scratch_elems = (long)NATOMS * 256 + NATOMS;   // summed + counts
    float* scratch;
    if (ws_size >= (size_t)scratch_elems * sizeof(float)) {
        scratch = (float*)d_ws;
    } else {
        // tail of out_edge region; consumed by kernel 2 before kernel 3 rewrites it
        scratch = out + (size_t)out_size - (size_t)scratch_elems;
    }
    float* summed = scratch;
    float* counts = scratch + (size_t)NATOMS * 256;

    const int zgrid = (int)((scratch_elems + 255) / 256);
    zero_kernel<<<zgrid, 256, 0, stream>>>(scratch, scratch_elems);

    const int egrid = (NENV + 127) / 128;                // 8 waves x 16 envs per block
    env_kernel<<<egrid, 256, 0, stream>>>(env_vectors, atom_attr, env_index,
                                          W1, b1, W2, b2, W3, b3, summed, counts);

    const int agrid = (NATOMS + 7) / 8;
    atom_kernel<<<agrid, 256, 0, stream>>>(summed, counts, g);

    edge_kernel<<<NEDGE, 256, 0, stream>>>(g, edge_index, edge_length, out_edge);
}